// ScaledDotProductAttention_22428319220810
// MI455X (gfx1250) — compile-verified
//
#include <hip/hip_runtime.h>

// ---------------------------------------------------------------------------
// Scaled dot-product attention (fp32, additive bias, materialized attention)
//   scores = (Q K^T) * 0.125 + prev ;  A = softmax(scores) ; O = A V
// Shapes: H=16, B=4, L=T=1024, D=64.  Memory-bound: ~600 MB @ 23.3 TB/s
// (~26 us floor) vs only ~17.5 GFLOP, so fp32 WMMA (v_wmma_f32_16x16x4_f32)
// matches reference numerics with compute to spare.
//
// Strategy: 32-row score stripe resident in LDS (~129 KB) so prev_attention
// is read once and attention written once; 129 KB allows TWO workgroups per
// 320 KB WGP -> 4 waves/SIMD for latency hiding on the streaming phases.
// prev loads / attention stores use non-temporal hints (read/written exactly
// once) so the 536 MB of streaming traffic does not thrash K/V out of L2.
// ---------------------------------------------------------------------------

typedef __attribute__((ext_vector_type(2))) float v2f;
typedef __attribute__((ext_vector_type(8))) float v8f;

namespace {
constexpr int kH = 16, kB = 4, kL = 1024, kT = 1024, kD = 64;
constexpr float kScale = 0.125f;
constexpr int kMRows = 32;              // rows of scores per workgroup
constexpr int kPitch = kT + 4;          // LDS row pitch (floats): +4 => row
                                        // stride of 4 banks; 16 row-strided
                                        // lanes hit 16 distinct banks.
}

__global__ __launch_bounds__(256) void
attn_bias_softmax_kernel(const float* __restrict__ Q,
                         const float* __restrict__ K,
                         const float* __restrict__ V,
                         const float* __restrict__ Pv,   // prev_attention
                         float* __restrict__ O,          // [H,B,L,D]
                         float* __restrict__ A)          // [H,B,L,T]
{
    extern __shared__ float s[];        // kMRows * kPitch floats (~129 KB)

    const int wg    = blockIdx.x;       // 2048 workgroups
    const int hb    = wg >> 5;          // (h,b) flat index, 64 of them
    const int mtile = wg & 31;          // which 32-row stripe of L
    const int row0  = mtile * kMRows;

    const float* Qp = Q  + (size_t)hb * kL * kD + (size_t)row0 * kD;
    const float* Kp = K  + (size_t)hb * kT * kD;
    const float* Vp = V  + (size_t)hb * kT * kD;
    const float* Pp = Pv + (size_t)hb * kL * kT + (size_t)row0 * kT;
    float*       Op = O  + (size_t)hb * kL * kD + (size_t)row0 * kD;
    float*       Ap = A  + (size_t)hb * kL * kT + (size_t)row0 * kT;

    const int lane = threadIdx.x & 31;
    const int wave = threadIdx.x >> 5;  // 8 waves
    const int l15  = lane & 15;
    const int hi   = lane >> 4;         // 0: lanes 0-15, 1: lanes 16-31

    // ---------------- Phase 1: raw scores = Q K^T into LDS ----------------
    // Wave grid: 2 row-stripes (16 rows each) x 4 column quarters (256 cols).
    // WMMA f32 16x16x4 A-frag layout: lane<16 holds K={d,d+1} of row lane,
    // lane>=16 holds K={d+2,d+3} of row lane-16 -> one float2 per step.
    {
        const int m0    = (wave & 1) * 16;
        const int nbase = (wave >> 1) * 256;

        v2f qa[16];                      // Q fragment, reused over 16 tiles
        #pragma unroll
        for (int kk = 0; kk < 16; ++kk) {
            const float2 t = *reinterpret_cast<const float2*>(
                Qp + (m0 + l15) * kD + kk * 4 + 2 * hi);
            qa[kk].x = t.x; qa[kk].y = t.y;
        }

        for (int nb = 0; nb < 16; ++nb) {
            const int n0 = nbase + nb * 16;
            v8f acc = {};
            #pragma unroll
            for (int kk = 0; kk < 16; ++kk) {
                // B-frag = K^T[4x16]: same per-lane float2 shape, rows of K.
                const float2 t = *reinterpret_cast<const float2*>(
                    Kp + (n0 + l15) * kD + kk * 4 + 2 * hi);
                v2f b; b.x = t.x; b.y = t.y;
                acc = __builtin_amdgcn_wmma_f32_16x16x4_f32(
                    false, qa[kk], false, b, (short)0, acc, false, false);
            }
            // D layout: VGPR j -> row m0+j (lanes 0-15) / m0+8+j (16-31).
            #pragma unroll
            for (int j = 0; j < 8; ++j)
                s[(m0 + j + 8 * hi) * kPitch + n0 + l15] = acc[j];
        }
    }
    __syncthreads();

    // -------- Phase 2: fuse scale + bias, softmax, emit attention ---------
    // Each wave owns 4 rows; prev_attention loads are fully coalesced,
    // touched exactly once, and non-temporal (268 MB streamed past L2).
    {
        for (int r = wave * 4; r < wave * 4 + 4; ++r) {
            float sv[32];
            float m = -3.402823466e+38f;
            #pragma unroll
            for (int i = 0; i < 32; ++i) {
                const int col = lane + i * 32;
                const float x = s[r * kPitch + col] * kScale
                    + __builtin_nontemporal_load(Pp + (size_t)r * kT + col);
                sv[i] = x;
                m = fmaxf(m, x);
            }
            #pragma unroll
            for (int off = 16; off > 0; off >>= 1)
                m = fmaxf(m, __shfl_xor(m, off, 32));

            float sum = 0.0f;
            #pragma unroll
            for (int i = 0; i < 32; ++i) {
                const float e = __expf(sv[i] - m);
                sv[i] = e;
                sum += e;
            }
            #pragma unroll
            for (int off = 16; off > 0; off >>= 1)
                sum += __shfl_xor(sum, off, 32);
            const float inv = 1.0f / sum;

            #pragma unroll
            for (int i = 0; i < 32; ++i) {
                const int col = lane + i * 32;
                const float p = sv[i] * inv;
                s[r * kPitch + col] = p;            // keep P for phase 3
                __builtin_nontemporal_store(p, Ap + (size_t)r * kT + col);
            }
        }
    }
    __syncthreads();

    // --------------------- Phase 3: O = P V (WMMA) ------------------------
    // 8 output tiles (2x4 of 16x16), 1 per wave. A-frag float2 from LDS:
    // row stride = 1028 floats = 4 banks => 16 row-strided lanes hit banks
    // {4r,4r+1}, upper half {4r+2,4r+3}: conflict-free ds_load_b64.
    // V is L2-resident (256 KB per (h,b) vs 192 MB L2).
    {
        const int m0 = (wave >> 2) * 16;
        const int n0 = (wave & 3) * 16;
        v8f acc = {};
        for (int d = 0; d < kT; d += 4) {
            const float2 pa = *reinterpret_cast<const float2*>(
                &s[(m0 + l15) * kPitch + d + 2 * hi]);
            v2f a; a.x = pa.x; a.y = pa.y;
            // B-frag rows of V: VGPR0 = row d+2*hi, VGPR1 = row d+1+2*hi.
            v2f b;
            b.x = Vp[(d + 2 * hi)     * kD + n0 + l15];
            b.y = Vp[(d + 2 * hi + 1) * kD + n0 + l15];
            acc = __builtin_amdgcn_wmma_f32_16x16x4_f32(
                false, a, false, b, (short)0, acc, false, false);
        }
        #pragma unroll
        for (int j = 0; j < 8; ++j)
            Op[(m0 + j + 8 * hi) * kD + n0 + l15] = acc[j];
    }
}

extern "C" void kernel_launch(void* const* d_in, const int* in_sizes, int n_in,
                              void* d_out, int out_size, void* d_ws, size_t ws_size,
                              hipStream_t stream) {
    (void)in_sizes; (void)n_in; (void)out_size; (void)d_ws; (void)ws_size;

    const float* Q  = (const float*)d_in[0];
    const float* K  = (const float*)d_in[1];
    const float* V  = (const float*)d_in[2];
    const float* Pv = (const float*)d_in[3];

    float* O = (float*)d_out;                               // [H,B,L,D] first
    float* A = (float*)d_out + (size_t)kH * kB * kL * kD;   // then [H,B,L,T]

    const size_t shmem = (size_t)kMRows * kPitch * sizeof(float); // ~129 KB
    // CDNA5 WGP allows up to 320 KB LDS; ~129 KB => 2 workgroups per WGP.
    hipFuncSetAttribute((const void*)attn_bias_softmax_kernel,
                        hipFuncAttributeMaxDynamicSharedMemorySize,
                        (int)shmem);

    const int wgs = kH * kB * (kL / kMRows);   // 64 * 32 = 2048 workgroups
    attn_bias_softmax_kernel<<<dim3(wgs), dim3(256), shmem, stream>>>(
        Q, K, V, Pv, O, A);
}